// E2AttentionArbOrder_sparse_9577777070591
// MI455X (gfx1250) — compile-verified
//
#include <hip/hip_runtime.h>
#include <hip/hip_bf16.h>

typedef __attribute__((ext_vector_type(2))) float v2f;
typedef __attribute__((ext_vector_type(8))) float v8f;

#define WMMA_F32X4(a, b, c) \
  __builtin_amdgcn_wmma_f32_16x16x4_f32(false, (a), false, (b), (short)0, (c), false, false)

__device__ __forceinline__ float sigmoidf_(float x) { return 1.0f / (1.0f + __expf(-x)); }
__device__ __forceinline__ float siluf_(float x)    { return x * sigmoidf_(x); }
__device__ __forceinline__ float slreluf_(float x)  { return 0.6f * x + 0.4f * x * (2.0f * sigmoidf_(x) - 1.0f); }

// row index within degree-l block -> (node, absolute m index in 0..8)
__device__ __forceinline__ void decode_row(int l, int r, int& n, int& mabs) {
  if (l == 0)      { n = r;     mabs = 0; }
  else if (l == 1) { n = r / 3; mabs = 1 + (r - n * 3); }
  else             { n = r / 5; mabs = 4 + (r - n * 5); }
}

// ---------------------------------------------------------------------------
// K1: dot = so3_linear(X, dot_W, dot_b)  -> Dot [N,9,32]   (WMMA f32 16x16x4)
// ---------------------------------------------------------------------------
__global__ __launch_bounds__(32) void dot_gemm_kernel(
    const float* __restrict__ X,     // [N,9,256]
    const float* __restrict__ W,     // [3,256,32]
    const float* __restrict__ bias,  // [32]
    float* __restrict__ Dot,         // [N,9,32]
    int N) {
  const int tiles0 = (N + 15) >> 4;
  const int tiles1 = (3 * N + 15) >> 4;
  int t = blockIdx.x;
  int l, rbase;
  if (t < tiles0)               { l = 0; rbase = t * 16; }
  else if (t < tiles0 + tiles1) { l = 1; rbase = (t - tiles0) * 16; }
  else                          { l = 2; rbase = (t - tiles0 - tiles1) * 16; }
  const int rows_l = (l == 0) ? N : (l == 1 ? 3 * N : 5 * N);

  const int lane = threadIdx.x;
  const int m16 = lane & 15;
  const int kh  = lane >> 4;  // 0: K pair {0,1}, 1: K pair {2,3}

  int rA = rbase + m16; if (rA >= rows_l) rA = rows_l - 1;
  int n, mabs; decode_row(l, rA, n, mabs);
  const float* arow = X + (size_t)(n * 9 + mabs) * 256;
  const float* Wl   = W + (size_t)l * 256 * 32;

  v8f c0 = {}; v8f c1 = {};
  for (int kk = 0; kk < 64; ++kk) {
    const int k0 = kk * 4 + kh * 2;
    v2f a;  a.x  = arow[k0];                 a.y  = arow[k0 + 1];
    v2f b0; b0.x = Wl[k0 * 32 + m16];        b0.y = Wl[(k0 + 1) * 32 + m16];
    v2f b1; b1.x = Wl[k0 * 32 + 16 + m16];   b1.y = Wl[(k0 + 1) * 32 + 16 + m16];
    c0 = WMMA_F32X4(a, b0, c0);
    c1 = WMMA_F32X4(a, b1, c1);
  }
#pragma unroll
  for (int i = 0; i < 8; ++i) {
    const int rD = rbase + i + kh * 8;
    if (rD >= rows_l) continue;
    int nn, mm; decode_row(l, rD, nn, mm);
    float v0 = c0[i], v1 = c1[i];
    if (l == 0) { v0 += bias[m16]; v1 += bias[16 + m16]; }
    float* o = Dot + (size_t)(nn * 9 + mm) * 32;
    o[m16] = v0; o[16 + m16] = v1;
  }
}

// ---------------------------------------------------------------------------
// K2: rotation invariants  Dot[N,9,32] -> Inv[N,96]
// ---------------------------------------------------------------------------
__global__ void invariants_kernel(const float* __restrict__ Dot,
                                  float* __restrict__ Inv, int N) {
  const int tid = blockIdx.x * blockDim.x + threadIdx.x;
  if (tid >= N * 32) return;
  const int n = tid >> 5, c = tid & 31;
  const float* d = Dot + (size_t)n * 9 * 32 + c;
  const float l0 = d[0];
  float s1 = 0.f, s2 = 0.f;
#pragma unroll
  for (int m = 1; m < 4; ++m) { float v = d[m * 32]; s1 += v * v; }
#pragma unroll
  for (int m = 4; m < 9; ++m) { float v = d[m * 32]; s2 += v * v; }
  float* o = Inv + (size_t)n * 96;
  o[c]      = l0;
  o[32 + c] = sqrtf(s1 + 1e-12f);
  o[64 + c] = sqrtf(s2 + 1e-12f);
}

// ---------------------------------------------------------------------------
// init kernels
// ---------------------------------------------------------------------------
__global__ void zero_f32_kernel(float* __restrict__ p, long long n) {
  long long i = (long long)blockIdx.x * blockDim.x + threadIdx.x;
  if (i < n) p[i] = 0.f;
}
__global__ void init_softmax_kernel(unsigned* __restrict__ amax,
                                    float* __restrict__ den, int n) {
  int i = blockIdx.x * blockDim.x + threadIdx.x;
  if (i < n) { amax[i] = 0u; den[i] = 0.f; }
}

// ---------------------------------------------------------------------------
// K3: per-edge radial MLPs + (pair*w_m0)@fc_m0_W (WMMA) + LN + SLReLU + logits
// One wave per 16 edges. Lanes 0-15: rad_m0; lanes 16-31: rad_v.
// ---------------------------------------------------------------------------
struct EdgeParams {
  const float *attn_w, *src_emb, *dst_emb;
  const int   *atomic_numbers, *edge_index;
  const float *W1m, *b1m, *g1m, *be1m, *W2m, *b2m, *g2m, *be2m, *W3m, *b3m;
  const float *W1v, *b1v, *g1v, *be1v, *W2v, *b2v, *g2v, *be2v, *W3v, *b3v;
  const float *fcW, *fcB, *lng, *lnb, *alpha, *inv;
  float *wv, *logits;
  int E;
};

__global__ __launch_bounds__(32) void edge_kernel(EdgeParams P) {
  __shared__ float Alds[16][200];   // (pair * w_m0) tile, [edge][192]
  __shared__ float aout[16][264];   // a = tile @ fc_m0_W, [edge][256]

  const int lane = threadIdx.x;
  const int m16 = lane & 15;
  const int kh  = lane >> 4;
  const int e0  = blockIdx.x * 16;
  int e = e0 + m16; if (e >= P.E) e = P.E - 1;

  const int s  = P.edge_index[e];
  const int d  = P.edge_index[P.E + e];
  const int as_ = P.atomic_numbers[s];
  const int ad_ = P.atomic_numbers[d];

  const bool isV = (kh == 1);
  const float* W1  = isV ? P.W1v  : P.W1m;
  const float* b1  = isV ? P.b1v  : P.b1m;
  const float* g1  = isV ? P.g1v  : P.g1m;
  const float* be1 = isV ? P.be1v : P.be1m;
  const float* W2  = isV ? P.W2v  : P.W2m;
  const float* b2  = isV ? P.b2v  : P.b2m;
  const float* g2  = isV ? P.g2v  : P.g2m;
  const float* be2 = isV ? P.be2v : P.be2m;

  // ---- layer 1: x_edge[288] @ W1[288,16] + b1, LN, SiLU ----
  float h1[16];
#pragma unroll
  for (int j = 0; j < 16; ++j) h1[j] = b1[j];
  for (int k = 0; k < 288; ++k) {
    float xk;
    if (k < 32)        xk = P.attn_w[(size_t)e * 32 + k];
    else if (k < 160)  xk = P.src_emb[(size_t)as_ * 128 + (k - 32)];
    else               xk = P.dst_emb[(size_t)ad_ * 128 + (k - 160)];
    const float* wr = W1 + (size_t)k * 16;
#pragma unroll
    for (int j = 0; j < 16; ++j) h1[j] += xk * wr[j];
  }
  {
    float mu = 0.f;
#pragma unroll
    for (int j = 0; j < 16; ++j) mu += h1[j];
    mu *= (1.f / 16.f);
    float var = 0.f;
#pragma unroll
    for (int j = 0; j < 16; ++j) { float t = h1[j] - mu; var += t * t; }
    var *= (1.f / 16.f);
    const float rstd = rsqrtf(var + 1e-5f);
#pragma unroll
    for (int j = 0; j < 16; ++j) h1[j] = siluf_((h1[j] - mu) * rstd * g1[j] + be1[j]);
  }

  // ---- layer 2: h1 @ W2[16,16] + b2, LN, SiLU ----
  float h2[16];
#pragma unroll
  for (int j = 0; j < 16; ++j) h2[j] = b2[j];
#pragma unroll
  for (int i = 0; i < 16; ++i) {
    const float* wr = W2 + (size_t)i * 16;
#pragma unroll
    for (int j = 0; j < 16; ++j) h2[j] += h1[i] * wr[j];
  }
  {
    float mu = 0.f;
#pragma unroll
    for (int j = 0; j < 16; ++j) mu += h2[j];
    mu *= (1.f / 16.f);
    float var = 0.f;
#pragma unroll
    for (int j = 0; j < 16; ++j) { float t = h2[j] - mu; var += t * t; }
    var *= (1.f / 16.f);
    const float rstd = rsqrtf(var + 1e-5f);
#pragma unroll
    for (int j = 0; j < 16; ++j) h2[j] = siluf_((h2[j] - mu) * rstd * g2[j] + be2[j]);
  }

  // ---- layer 3 + fused gather ----
  if (!isV) {
    // w_m0[c] * pair[c] -> LDS A tile
    for (int c = 0; c < 192; ++c) {
      float w = P.b3m[c];
#pragma unroll
      for (int j = 0; j < 16; ++j) w += h2[j] * P.W3m[(size_t)j * 192 + c];
      const float pc = (c < 96) ? P.inv[(size_t)s * 96 + c]
                                : P.inv[(size_t)d * 96 + (c - 96)];
      Alds[m16][c] = pc * w;
    }
  } else {
    // wv[c] -> global (needed later by the scatter kernel)
    for (int c = 0; c < 256; ++c) {
      float w = P.b3v[c];
#pragma unroll
      for (int j = 0; j < 16; ++j) w += h2[j] * P.W3v[(size_t)j * 256 + c];
      P.wv[(size_t)e * 256 + c] = w;
    }
  }
  __syncthreads();

  // ---- WMMA: [16,192] @ fc_m0_W[192,256] + fc_m0_b -> aout ----
  for (int ct = 0; ct < 16; ++ct) {
    const int col = ct * 16 + m16;
    v8f c = {};
    for (int kk = 0; kk < 48; ++kk) {
      const int k0 = kk * 4 + kh * 2;
      v2f a; a.x = Alds[m16][k0];                  a.y = Alds[m16][k0 + 1];
      v2f b; b.x = P.fcW[(size_t)k0 * 256 + col];  b.y = P.fcW[(size_t)(k0 + 1) * 256 + col];
      c = WMMA_F32X4(a, b, c);
    }
    const float bc = P.fcB[col];
#pragma unroll
    for (int i = 0; i < 8; ++i) aout[i + kh * 8][col] = c[i] + bc;
  }
  __syncthreads();

  // ---- per-(edge,head): LN over S=32, smooth-leaky, alpha-dot -> logits ----
  // lane handles edge m16, heads [kh*4, kh*4+4)
  const int ee = e0 + m16;
#pragma unroll
  for (int hh = 0; hh < 4; ++hh) {
    const int h = kh * 4 + hh;
    const float* row = &aout[m16][h * 32];
    float mu = 0.f;
#pragma unroll
    for (int s2 = 0; s2 < 32; ++s2) mu += row[s2];
    mu *= (1.f / 32.f);
    float var = 0.f;
#pragma unroll
    for (int s2 = 0; s2 < 32; ++s2) { float t = row[s2] - mu; var += t * t; }
    var *= (1.f / 32.f);
    const float rstd = rsqrtf(var + 1e-5f);
    float acc = 0.f;
#pragma unroll
    for (int s2 = 0; s2 < 32; ++s2) {
      const float v = (row[s2] - mu) * rstd * P.lng[s2] + P.lnb[s2];
      acc += slreluf_(v) * P.alpha[h * 32 + s2];
    }
    if (ee < P.E) P.logits[(size_t)ee * 8 + h] = acc;
  }
}

// ---------------------------------------------------------------------------
// segment softmax over dst
// ---------------------------------------------------------------------------
__device__ __forceinline__ unsigned f2ord(float f) {
  unsigned b = __float_as_uint(f);
  return (b & 0x80000000u) ? ~b : (b | 0x80000000u);
}
__device__ __forceinline__ float ord2f(unsigned u) {
  unsigned b = (u & 0x80000000u) ? (u & 0x7FFFFFFFu) : ~u;
  return __uint_as_float(b);
}

__global__ void logit_max_kernel(const float* __restrict__ logits,
                                 const int* __restrict__ edge_index,
                                 unsigned* __restrict__ amax, int E) {
  int t = blockIdx.x * blockDim.x + threadIdx.x;
  if (t >= E * 8) return;
  const int e = t >> 3, h = t & 7;
  const int d = edge_index[E + e];
  atomicMax(&amax[d * 8 + h], f2ord(logits[t]));
}

__global__ void exp_kernel(const float* __restrict__ logits,
                           const int* __restrict__ edge_index,
                           const unsigned* __restrict__ amax,
                           float* __restrict__ ex, float* __restrict__ den, int E) {
  int t = blockIdx.x * blockDim.x + threadIdx.x;
  if (t >= E * 8) return;
  const int e = t >> 3, h = t & 7;
  const int d = edge_index[E + e];
  float am = ord2f(amax[d * 8 + h]);
  if (!(am > -3.0e38f && am < 3.0e38f)) am = 0.f;  // isfinite guard
  const float v = __expf(logits[t] - am);
  ex[t] = v;
  atomicAdd(&den[d * 8 + h], v);
}

__global__ void attn_kernel(float* __restrict__ ex,
                            const int* __restrict__ edge_index,
                            const float* __restrict__ den, int E) {
  int t = blockIdx.x * blockDim.x + threadIdx.x;
  if (t >= E * 8) return;
  const int e = t >> 3, h = t & 7;
  const int d = edge_index[E + e];
  ex[t] = ex[t] / (den[d * 8 + h] + 1e-9f);
}

// ---------------------------------------------------------------------------
// K5: msg scatter: acc[dst,m,c] += X[src,m,c] * wv[e,c] * attn[e, c/32]
// ---------------------------------------------------------------------------
__global__ void scatter_kernel(const float* __restrict__ X,
                               const float* __restrict__ wv,
                               const float* __restrict__ attn,
                               const int* __restrict__ edge_index,
                               float* __restrict__ acc, int E) {
  int t = blockIdx.x * blockDim.x + threadIdx.x;
  if (t >= E * 256) return;
  const int e = t >> 8, c = t & 255;
  const int s = edge_index[e];
  const int d = edge_index[E + e];
  const float g = wv[t] * attn[e * 8 + (c >> 5)];
  const float* xs = X + (size_t)s * 9 * 256 + c;
  float* ac = acc + (size_t)d * 9 * 256 + c;
#pragma unroll
  for (int m = 0; m < 9; ++m) atomicAdd(ac + m * 256, xs[m * 256] * g);
}

// ---------------------------------------------------------------------------
// K6: out = so3_linear(acc, proj_W, proj_b)  (WMMA f32 16x16x4, LDS-staged A)
// ---------------------------------------------------------------------------
__global__ __launch_bounds__(32) void proj_gemm_kernel(
    const float* __restrict__ Acc,   // [N,9,256]
    const float* __restrict__ W,     // [3,256,256]
    const float* __restrict__ bias,  // [256]
    float* __restrict__ Out,         // [N,9,256]
    int N) {
  __shared__ float Alds[16][260];

  const int tiles0 = (N + 15) >> 4;
  const int tiles1 = (3 * N + 15) >> 4;
  int t = blockIdx.x;
  int l, rbase;
  if (t < tiles0)               { l = 0; rbase = t * 16; }
  else if (t < tiles0 + tiles1) { l = 1; rbase = (t - tiles0) * 16; }
  else                          { l = 2; rbase = (t - tiles0 - tiles1) * 16; }
  const int rows_l = (l == 0) ? N : (l == 1 ? 3 * N : 5 * N);

  const int lane = threadIdx.x;
  // stage A tile: 16 rows x 256 cols
  for (int idx = lane; idx < 16 * 256; idx += 32) {
    const int r = idx >> 8, k = idx & 255;
    int rr = rbase + r; if (rr >= rows_l) rr = rows_l - 1;
    int n, m; decode_row(l, rr, n, m);
    Alds[r][k] = Acc[(size_t)(n * 9 + m) * 256 + k];
  }
  __syncthreads();

  const float* Wl = W + (size_t)l * 256 * 256;
  const int m16 = lane & 15;
  const int kh  = lane >> 4;

  for (int ct = 0; ct < 16; ++ct) {
    const int col = ct * 16 + m16;
    v8f c = {};
    for (int kk = 0; kk < 64; ++kk) {
      const int k0 = kk * 4 + kh * 2;
      v2f a; a.x = Alds[m16][k0];                a.y = Alds[m16][k0 + 1];
      v2f b; b.x = Wl[(size_t)k0 * 256 + col];   b.y = Wl[(size_t)(k0 + 1) * 256 + col];
      c = WMMA_F32X4(a, b, c);
    }
    const float bc = (l == 0) ? bias[col] : 0.f;
#pragma unroll
    for (int i = 0; i < 8; ++i) {
      const int rD = rbase + i + kh * 8;
      if (rD >= rows_l) continue;
      int nn, mm; decode_row(l, rD, nn, mm);
      Out[(size_t)(nn * 9 + mm) * 256 + col] = c[i] + bc;
    }
  }
}

// ---------------------------------------------------------------------------
// host launch
// ---------------------------------------------------------------------------
extern "C" void kernel_launch(void* const* d_in, const int* in_sizes, int n_in,
                              void* d_out, int out_size, void* d_ws, size_t ws_size,
                              hipStream_t stream) {
  // setup_inputs() insertion order, params flattened recursively:
  const float* X           = (const float*)d_in[1];   // node_irreps_input [N,9,256]
  const float* attn_w      = (const float*)d_in[4];   // [E,32]
  const int*   atomic_nums = (const int*)  d_in[5];   // [N]
  const int*   edge_index  = (const int*)  d_in[6];   // [2,E]
  const float* src_emb     = (const float*)d_in[7];
  const float* dst_emb     = (const float*)d_in[8];
  const float* dot_W       = (const float*)d_in[9];
  const float* dot_b       = (const float*)d_in[10];
  // rad_m0: W1,b1,g1,be1,W2,b2,g2,be2,W3,b3 = d_in[11..20]
  const float* m0W1 = (const float*)d_in[11]; const float* m0b1 = (const float*)d_in[12];
  const float* m0g1 = (const float*)d_in[13]; const float* m0be1= (const float*)d_in[14];
  const float* m0W2 = (const float*)d_in[15]; const float* m0b2 = (const float*)d_in[16];
  const float* m0g2 = (const float*)d_in[17]; const float* m0be2= (const float*)d_in[18];
  const float* m0W3 = (const float*)d_in[19]; const float* m0b3 = (const float*)d_in[20];
  const float* fcW   = (const float*)d_in[21];
  const float* fcB   = (const float*)d_in[22];
  const float* lng   = (const float*)d_in[23];
  const float* lnb   = (const float*)d_in[24];
  const float* alpha = (const float*)d_in[25];
  // rad_v: d_in[26..35]
  const float* vW1 = (const float*)d_in[26]; const float* vb1 = (const float*)d_in[27];
  const float* vg1 = (const float*)d_in[28]; const float* vbe1= (const float*)d_in[29];
  const float* vW2 = (const float*)d_in[30]; const float* vb2 = (const float*)d_in[31];
  const float* vg2 = (const float*)d_in[32]; const float* vbe2= (const float*)d_in[33];
  const float* vW3 = (const float*)d_in[34]; const float* vb3 = (const float*)d_in[35];
  const float* proj_W = (const float*)d_in[36];
  const float* proj_b = (const float*)d_in[37];

  const int N = in_sizes[5];  // atomic_numbers
  const int E = in_sizes[2];  // edge_dis

  // carve workspace
  char* w = (char*)d_ws;
  auto carve = [&](size_t bytes) { void* p = (void*)w; w += (bytes + 255) & ~(size_t)255; return p; };
  float*    d_dot    = (float*)   carve((size_t)N * 9 * 32 * 4);
  float*    d_inv    = (float*)   carve((size_t)N * 96 * 4);
  float*    d_wv     = (float*)   carve((size_t)E * 256 * 4);
  float*    d_logits = (float*)   carve((size_t)E * 8 * 4);
  float*    d_attn   = (float*)   carve((size_t)E * 8 * 4);
  unsigned* d_amax   = (unsigned*)carve((size_t)N * 8 * 4);
  float*    d_den    = (float*)   carve((size_t)N * 8 * 4);
  float*    d_acc    = (float*)   carve((size_t)N * 9 * 256 * 4);

  const int rowTiles = ((N + 15) >> 4) + ((3 * N + 15) >> 4) + ((5 * N + 15) >> 4);

  // K1: dot GEMM
  dot_gemm_kernel<<<rowTiles, 32, 0, stream>>>(X, dot_W, dot_b, d_dot, N);
  // K2: invariants
  invariants_kernel<<<(N * 32 + 255) / 256, 256, 0, stream>>>(d_dot, d_inv, N);
  // init accumulators
  {
    long long accN = (long long)N * 9 * 256;
    zero_f32_kernel<<<(unsigned)((accN + 255) / 256), 256, 0, stream>>>(d_acc, accN);
    init_softmax_kernel<<<(N * 8 + 255) / 256, 256, 0, stream>>>(d_amax, d_den, N * 8);
  }
  // K3: edge pipeline
  {
    EdgeParams P;
    P.attn_w = attn_w; P.src_emb = src_emb; P.dst_emb = dst_emb;
    P.atomic_numbers = atomic_nums; P.edge_index = edge_index;
    P.W1m = m0W1; P.b1m = m0b1; P.g1m = m0g1; P.be1m = m0be1;
    P.W2m = m0W2; P.b2m = m0b2; P.g2m = m0g2; P.be2m = m0be2;
    P.W3m = m0W3; P.b3m = m0b3;
    P.W1v = vW1;  P.b1v = vb1;  P.g1v = vg1;  P.be1v = vbe1;
    P.W2v = vW2;  P.b2v = vb2;  P.g2v = vg2;  P.be2v = vbe2;
    P.W3v = vW3;  P.b3v = vb3;
    P.fcW = fcW; P.fcB = fcB; P.lng = lng; P.lnb = lnb; P.alpha = alpha;
    P.inv = d_inv; P.wv = d_wv; P.logits = d_logits; P.E = E;
    edge_kernel<<<(E + 15) / 16, 32, 0, stream>>>(P);
  }
  // K4: segment softmax
  logit_max_kernel<<<(E * 8 + 255) / 256, 256, 0, stream>>>(d_logits, edge_index, d_amax, E);
  exp_kernel<<<(E * 8 + 255) / 256, 256, 0, stream>>>(d_logits, edge_index, d_amax, d_attn, d_den, E);
  attn_kernel<<<(E * 8 + 255) / 256, 256, 0, stream>>>(d_attn, edge_index, d_den, E);
  // K5: scatter messages
  scatter_kernel<<<(E * 256 + 255) / 256, 256, 0, stream>>>(X, d_wv, d_attn, edge_index, d_acc, E);
  // K6: output projection GEMM
  proj_gemm_kernel<<<rowTiles, 32, 0, stream>>>(d_acc, proj_W, proj_b, (float*)d_out, N);
}